// FSIM_5549097746780
// MI455X (gfx1250) — compile-verified
//
#include <hip/hip_runtime.h>
#include <hip/hip_bf16.h>

// ---------------------------------------------------------------------------
// FSIM for MI455X (gfx1250). DFT-as-GEMM on V_WMMA_F32_16X16X4_F32 with
// async global->LDS staging of the shared A strip (CDNA5 ASYNCcnt path).
// All matrices are 256x256 row-major. Complex planes are stored as
// [real(65536) | imag(65536)] contiguous floats.
// ---------------------------------------------------------------------------

typedef __attribute__((ext_vector_type(2))) float v2f;
typedef __attribute__((ext_vector_type(8))) float v8f;

#define PIXELS 65536   // 256*256
#define DIM 256
#define LROW 260       // padded LDS row stride (floats): 4*m mod 64 -> no bank conflicts
static __device__ __constant__ float PI_F = 3.14159265358979323846f;

// CDNA5 async copy: 16 bytes global -> LDS per lane (tracked by ASYNCcnt).
__device__ __forceinline__ void async_g2l_16B(unsigned lds_byte_off, const float* g)
{
    unsigned long long ga = (unsigned long long)(const void*)g;
    asm volatile("global_load_async_to_lds_b128 %0, %1, off"
                 :: "v"(lds_byte_off), "v"(ga) : "memory");
}
__device__ __forceinline__ void wait_async0()
{
    asm volatile("s_wait_asynccnt 0" ::: "memory");
}

// ---------------------------------------------------------------------------
// Batched complex GEMM: C[z] = scale * (A[z] (.* filt[z])?) * B[z]
//   A: real plane at A + z*aZstr, imag (COMPLEX_A) at +aImagOff; true imag
//      value = aISign * stored (conj via -1). filt: optional real multiplier.
//   B: complex, imag at +bImagOff, true imag = bISign * stored.
//   C: complex out, imag at +65536.
// One wave computes one 16x16 tile; the 8 waves of a block share one 16-row
// A strip, staged once into LDS via async copies. Signs are applied in the
// epilogue so the K-loop is pure LDS/global loads + 4 WMMAs.
// ---------------------------------------------------------------------------
template<bool COMPLEX_A, bool USE_FILT>
__global__ void __launch_bounds__(256)
k_cgemm(const float* __restrict__ A, int aZstr, int aImagOff, float aISign,
        const float* __restrict__ filt, int filtZstr,
        const float* __restrict__ B, int bZstr, int bImagOff, float bISign,
        float* __restrict__ C, int cZstr, float scale)
{
    constexpr int STRIPF = 16 * LROW;                    // padded strip (floats)
    constexpr int NBUF   = 1 + (COMPLEX_A ? 1 : 0) + (USE_FILT ? 1 : 0);
    __shared__ __align__(16) float lds[STRIPF * NBUF];
    float* lAr = lds;
    float* lAi = COMPLEX_A ? lds + STRIPF : lds;
    float* lF  = USE_FILT  ? lds + STRIPF * (COMPLEX_A ? 2 : 1) : lds;

    const int z    = blockIdx.z;
    const int lane = threadIdx.x & 31;
    const int wave = threadIdx.x >> 5;
    const int tm   = (blockIdx.x >> 1) << 4;             // block-uniform row base
    const int tn   = (((blockIdx.x & 1) << 3) + wave) << 4;
    const int m    = lane & 15;
    const int kb   = (lane >> 4) << 1;                   // 0 or 2

    const float* Az = A + (size_t)z * aZstr + tm * DIM;  // 16-row strip, contiguous
    const float* Fz = USE_FILT ? (filt + (size_t)z * filtZstr + tm * DIM) : nullptr;
    const float* Bz = B + (size_t)z * bZstr;

    // ---- cooperative async stage of the A strip (+imag, +filter) ----
    {
        const unsigned lr = (unsigned)(unsigned long long)(const void*)lAr;
        const unsigned li = (unsigned)(unsigned long long)(const void*)lAi;
        const unsigned lf = (unsigned)(unsigned long long)(const void*)lF;
        for (int c = threadIdx.x; c < 1024; c += 256) {  // 1024 chunks of 16B
            const int e    = c << 2;                     // logical float index
            const int row  = e >> 8, col = e & 255;
            const unsigned lo = (unsigned)(row * LROW + col) << 2;
            async_g2l_16B(lr + lo, Az + e);
            if (COMPLEX_A) async_g2l_16B(li + lo, Az + aImagOff + e);
            if (USE_FILT)  async_g2l_16B(lf + lo, Fz + e);
        }
        wait_async0();
        __syncthreads();
        if (USE_FILT) {                                  // fuse filter once in LDS
            for (int e = threadIdx.x; e < 4096; e += 256) {
                const int li2 = (e >> 8) * LROW + (e & 255);
                lAr[li2] *= lF[li2];
                if (COMPLEX_A) lAi[li2] *= lF[li2];
            }
            __syncthreads();
        }
    }

    v8f p  = {};   // sum Ar*Br
    v8f q  = {};   // sum Ai*Bi   (COMPLEX_A)
    v8f r1 = {};   // sum Ar*Bi
    v8f r2 = {};   // sum Ai*Br   (COMPLEX_A)

    #pragma unroll 4
    for (int kk = 0; kk < DIM; kk += 4) {
        const int ka = kk + kb;
        v2f ar = *(const v2f*)&lAr[m * LROW + ka];
        const float* bcol = Bz + (size_t)ka * DIM + tn + m;
        __builtin_prefetch(bcol + 4 * DIM, 0, 1);        // global_prefetch_b8
        v2f br; br.x = bcol[0];        br.y = bcol[DIM];
        v2f bi; bi.x = bcol[bImagOff]; bi.y = bcol[bImagOff + DIM];
        p  = __builtin_amdgcn_wmma_f32_16x16x4_f32(false, ar, false, br, (short)0, p,  false, false);
        r1 = __builtin_amdgcn_wmma_f32_16x16x4_f32(false, ar, false, bi, (short)0, r1, false, false);
        if (COMPLEX_A) {
            v2f ai = *(const v2f*)&lAi[m * LROW + ka];
            q  = __builtin_amdgcn_wmma_f32_16x16x4_f32(false, ai, false, bi, (short)0, q,  false, false);
            r2 = __builtin_amdgcn_wmma_f32_16x16x4_f32(false, ai, false, br, (short)0, r2, false, false);
        }
    }

    // Epilogue: apply conjugation signs once.
    const float st = aISign * bISign;
    float* Cz = C + (size_t)z * cZstr;
    const int rbase = tm + ((lane >> 4) << 3);
    const int col   = tn + (lane & 15);
    #pragma unroll
    for (int j = 0; j < 8; ++j) {
        const int row = rbase + j;
        const float cr = COMPLEX_A ? (p[j] - st * q[j]) : p[j];
        const float cm = COMPLEX_A ? (bISign * r1[j] + aISign * r2[j]) : (bISign * r1[j]);
        Cz[row * DIM + col]          = scale * cr;
        Cz[PIXELS + row * DIM + col] = scale * cm;
    }
}

// ---------------------------------------------------------------------------
// 2x2 downsample + RGB->YIQ. Output layout: [side(2)][n(16)][d(3)][65536]
// ---------------------------------------------------------------------------
__global__ void __launch_bounds__(256)
k_yiq(const float* __restrict__ x, const float* __restrict__ y, float* __restrict__ yiq)
{
    const float M[9] = { 0.299f, 0.587f, 0.114f,
                         0.5969f, -0.2746f, -0.3213f,
                         0.2115f, -0.5227f, 0.3112f };
    size_t tid = (size_t)blockIdx.x * 256 + threadIdx.x;   // 96*65536 total
    int idx   = (int)(tid & (PIXELS - 1));
    int plane = (int)(tid >> 16);
    int d    = plane % 3;
    int n    = (plane / 3) % 16;
    int side = plane / 48;
    const float* base = (side == 0 ? x : y) + (size_t)n * 3 * 512 * 512;
    int i = idx >> 8, j = idx & 255;
    float acc = 0.0f;
    for (int c = 0; c < 3; ++c) {
        const float* p = base + (size_t)c * 512 * 512;
        float s = p[(2*i)*512 + 2*j]     + p[(2*i)*512 + 2*j + 1]
                + p[(2*i+1)*512 + 2*j]   + p[(2*i+1)*512 + 2*j + 1];
        acc += M[d*3 + c] * (s * 0.25f);
    }
    yiq[tid] = acc;
}

// DFT matrix F[k][n] = exp(-2*pi*i*k*n/256), stored [Re|Im].
__global__ void __launch_bounds__(256) k_dft(float* __restrict__ F)
{
    int idx = blockIdx.x * 256 + threadIdx.x;
    int k = idx >> 8, n = idx & 255;
    float ang = -2.0f * PI_F * (float)((k * n) & 255) / 256.0f;
    float s, c; __sincosf(ang, &s, &c);
    F[idx] = c;
    F[PIXELS + idx] = s;
}

__global__ void k_zero(float* p, int n)
{ int i = blockIdx.x * blockDim.x + threadIdx.x; if (i < n) p[i] = 0.0f; }
__global__ void k_zero_u32(unsigned* p, int n)
{ int i = blockIdx.x * blockDim.x + threadIdx.x; if (i < n) p[i] = 0u; }

// ---------------------------------------------------------------------------
// Log-Gabor filter bank: filt[s*4+o][65536] ; em[o] += filt[0][o]^2
// ---------------------------------------------------------------------------
__global__ void __launch_bounds__(256)
k_filters(float* __restrict__ filt, float* __restrict__ em)
{
    int idx = blockIdx.x * 256 + threadIdx.x;
    int i = idx >> 8, j = idx & 255;
    float u = (float)((i < 128) ? i : i - 256) / 256.0f;   // ifftshifted freq
    float v = (float)((j < 128) ? j : j - 256) / 256.0f;
    float r = sqrtf(u*u + v*v);
    float theta = atan2f(-v, u);
    float lowpass = 1.0f / (1.0f + powf(r / 0.45f, 30.0f));
    float rs = (idx == 0) ? 1.0f : r;
    float lnsf = logf(0.5978f);
    float denomA = 2.0f * lnsf * lnsf;
    float st = sinf(theta), ct = cosf(theta);
    for (int o = 0; o < 4; ++o) {
        float tj = PI_F * (float)o / 4.0f;
        float sj, cj; __sincosf(tj, &sj, &cj);
        float dt = atan2f(st * cj - ct * sj, ct * cj + st * sj);
        float b = expf(-dt*dt / (2.0f * 0.6545f * 0.6545f));
        for (int s = 0; s < 4; ++s) {
            float f0 = 1.0f / (6.0f * exp2f((float)s));
            float lr = logf(rs / f0);
            float a = expf(-lr*lr / denomA) * lowpass;
            if (idx == 0) a = 0.0f;
            float val = a * b;
            filt[(s*4 + o) * PIXELS + idx] = val;
            if (s == 0) atomicAdd(&em[o], val * val);
        }
    }
}

// sum_aiaj[o] += ( sum_s Re(ifft2(filt[s][o])) )^2 over pixels  (x HW later)
__global__ void __launch_bounds__(256)
k_red_g(const float* __restrict__ eo, float* __restrict__ saa)
{
    int idx = blockIdx.x * 256 + threadIdx.x;
    int o = blockIdx.y;
    float g = 0.0f;
    for (int s = 0; s < 4; ++s) g += eo[(size_t)(s*4 + o) * 2 * PIXELS + idx];
    float val = g * g;
    __shared__ float sh[256];
    sh[threadIdx.x] = val; __syncthreads();
    for (int off = 128; off; off >>= 1) {
        if (threadIdx.x < (unsigned)off) sh[threadIdx.x] += sh[threadIdx.x + off];
        __syncthreads();
    }
    if (threadIdx.x == 0) atomicAdd(&saa[o], sh[0]);
}

// ---------------------------------------------------------------------------
// Per-image energy / e2 / amplitude-sum from the 16 eo planes.
// ---------------------------------------------------------------------------
__global__ void __launch_bounds__(256)
k_energy(const float* __restrict__ eo, float* __restrict__ energy,
         float* __restrict__ e2, float* __restrict__ ampsum)
{
    int idx = blockIdx.x * 256 + threadIdx.x;
    float amp_tot = 0.0f;
    for (int o = 0; o < 4; ++o) {
        float sr = 0.0f, si = 0.0f, er[4], ei[4];
        for (int s = 0; s < 4; ++s) {
            er[s] = eo[(size_t)(s*4 + o) * 2 * PIXELS + idx];
            ei[s] = eo[(size_t)(s*4 + o) * 2 * PIXELS + PIXELS + idx];
            sr += er[s]; si += ei[s];
        }
        float an = sqrtf(sr*sr + si*si) + 1e-8f;
        float ur = sr / an, ui = si / an;
        float e = 0.0f;
        for (int s = 0; s < 4; ++s) {
            float re = er[s]*ur + ei[s]*ui;
            float im = ei[s]*ur - er[s]*ui;
            e += re - fabsf(im);
            amp_tot += sqrtf(er[s]*er[s] + ei[s]*ei[s]);
        }
        energy[(size_t)o * PIXELS + idx] = e;
        e2[(size_t)o * PIXELS + idx] = er[0]*er[0] + ei[0]*ei[0];
    }
    ampsum[idx] = amp_tot;
}

// ---------------------------------------------------------------------------
// Radix-select median (rank 32767) over 128 groups of 65536 non-neg floats.
// ---------------------------------------------------------------------------
__global__ void k_ms_init(unsigned* prefix, int* rank)
{ int g = threadIdx.x; if (g < 128) { prefix[g] = 0u; rank[g] = 32767; } }

__global__ void __launch_bounds__(256)
k_hist(const float* __restrict__ e2, const unsigned* __restrict__ prefix,
       unsigned* __restrict__ hist, int shift)
{
    int g = blockIdx.y;
    __shared__ unsigned sh[256];
    sh[threadIdx.x] = 0u; __syncthreads();
    unsigned pre = prefix[g];
    unsigned maskHi = (shift == 24) ? 0u : (0xFFFFFFFFu << (shift + 8));
    const float* src = e2 + (size_t)g * PIXELS;
    for (int i = blockIdx.x * 256 + threadIdx.x; i < PIXELS; i += gridDim.x * 256) {
        unsigned u = __float_as_uint(src[i]);
        if ((u & maskHi) == pre) atomicAdd(&sh[(u >> shift) & 255], 1u);
    }
    __syncthreads();
    if (sh[threadIdx.x]) atomicAdd(&hist[g * 256 + threadIdx.x], sh[threadIdx.x]);
}

__global__ void k_select(unsigned* prefix, int* rank,
                         const unsigned* __restrict__ hist, int shift)
{
    int g = threadIdx.x; if (g >= 128) return;
    int r = rank[g];
    unsigned cum = 0; int bin = 255;
    for (int b = 0; b < 256; ++b) {
        unsigned c = hist[g * 256 + b];
        if (cum + c > (unsigned)r) { bin = b; break; }
        cum += c;
    }
    prefix[g] |= (unsigned)bin << shift;
    rank[g] = r - (int)cum;
}

__global__ void k_thresh(const unsigned* __restrict__ prefix,
                         const float* __restrict__ em,
                         const float* __restrict__ saa, float* __restrict__ thr)
{
    int g = threadIdx.x; if (g >= 128) return;
    int o = g & 3;
    float med = __uint_as_float(prefix[g]);
    float mean_e2 = -med / logf(0.5f);
    float npow = mean_e2 / em[o];
    float tau = sqrtf(npow * saa[o] * (float)PIXELS);
    float c = sqrtf(PI_F * 0.5f), d = sqrtf(2.0f - PI_F * 0.5f);
    thr[g] = tau * (c + 2.0f * d) / 1.7f;
}

// Scharr gradient magnitude on the Y channel of each of the 32 images.
__global__ void __launch_bounds__(256)
k_scharr(const float* __restrict__ yiq, const float* __restrict__ kern,
         float* __restrict__ gmag)
{
    int z = blockIdx.z;
    int idx = blockIdx.x * 256 + threadIdx.x;
    int i = idx >> 8, j = idx & 255;
    const float* src = yiq + (size_t)z * 3 * PIXELS;   // channel 0 (Y)
    float gh = 0.0f, gv = 0.0f;
    for (int di = -1; di <= 1; ++di)
        for (int dj = -1; dj <= 1; ++dj) {
            int ii = i + di, jj = j + dj;
            float v = (ii >= 0 && ii < DIM && jj >= 0 && jj < DIM) ? src[ii * DIM + jj] : 0.0f;
            gh += kern[(di + 1) * 3 + (dj + 1)] * v;
            gv += kern[9 + (di + 1) * 3 + (dj + 1)] * v;
        }
    gmag[(size_t)z * PIXELS + idx] = sqrtf(gh * gh + gv * gv);
}

// Final similarity map + weighted reduction per image.
__global__ void __launch_bounds__(256)
k_fsacc(const float* __restrict__ energy, const float* __restrict__ thr,
        const float* __restrict__ ampsum, const float* __restrict__ gmag,
        const float* __restrict__ yiq, float* __restrict__ num, float* __restrict__ den)
{
    int img = blockIdx.z;          // 0..15
    int idx = blockIdx.x * 256 + threadIdx.x;

    auto pcf = [&](int z) -> float {
        float a = ampsum[(size_t)z * PIXELS + idx] + 1e-8f;
        float e = 0.0f;
        for (int o = 0; o < 4; ++o)
            e += fmaxf(energy[((size_t)z * 4 + o) * PIXELS + idx] - thr[z * 4 + o], 0.0f);
        return e / a;
    };

    float pcx = pcf(img), pcy = pcf(16 + img);
    const float t1 = 0.85f;
    float s_pc = (2.0f * pcx * pcy + t1) / (pcx * pcx + pcy * pcy + t1);

    float gx = gmag[(size_t)img * PIXELS + idx];
    float gy = gmag[(size_t)(16 + img) * PIXELS + idx];
    const float t2 = 160.0f / (255.0f * 255.0f);
    float s_g = (2.0f * gx * gy + t2) / (gx * gx + gy * gy + t2);

    const float* yx = yiq + (size_t)img * 3 * PIXELS;
    const float* yy = yiq + (size_t)(16 + img) * 3 * PIXELS;
    float ix = yx[PIXELS + idx],     iy = yy[PIXELS + idx];
    float qx = yx[2 * PIXELS + idx], qy = yy[2 * PIXELS + idx];
    const float t3 = 200.0f / (255.0f * 255.0f);
    float s_i = (2.0f * ix * iy + t3) / (ix * ix + iy * iy + t3);
    float s_q = (2.0f * qx * qy + t3) / (qx * qx + qy * qy + t3);
    float siq = s_i * s_q;
    const float lmb = 0.03f;
    float siql = powf(fabsf(siq), lmb) * ((siq < 0.0f) ? cosf(lmb * PI_F) : 1.0f);

    float s_l = s_pc * s_g * siql;
    float pcm = fmaxf(pcx, pcy);

    __shared__ float sn[256], sd[256];
    sn[threadIdx.x] = s_l * pcm; sd[threadIdx.x] = pcm; __syncthreads();
    for (int off = 128; off; off >>= 1) {
        if (threadIdx.x < (unsigned)off) {
            sn[threadIdx.x] += sn[threadIdx.x + off];
            sd[threadIdx.x] += sd[threadIdx.x + off];
        }
        __syncthreads();
    }
    if (threadIdx.x == 0) { atomicAdd(&num[img], sn[0]); atomicAdd(&den[img], sd[0]); }
}

__global__ void k_finish(const float* num, const float* den, float* out)
{
    if (threadIdx.x == 0) {
        float s = 0.0f;
        for (int i = 0; i < 16; ++i) s += num[i] / den[i];
        out[0] = s / 16.0f;
    }
}

// ---------------------------------------------------------------------------
extern "C" void kernel_launch(void* const* d_in, const int* in_sizes, int n_in,
                              void* d_out, int out_size, void* d_ws, size_t ws_size,
                              hipStream_t stream)
{
    (void)in_sizes; (void)n_in; (void)out_size; (void)ws_size;
    const float* x    = (const float*)d_in[0];
    const float* y    = (const float*)d_in[1];
    const float* kern = (const float*)d_in[2];
    float* out = (float*)d_out;

    const size_t P = PIXELS;
    float* w      = (float*)d_ws;
    float* yiq    = w;                    //  96*P   [side][n][d]
    float* F      = yiq   + 96 * P;       //   2*P   DFT matrix (Re|Im)
    float* filt   = F     + 2 * P;        //  16*P   filter bank
    float* Xf     = filt  + 16 * P;       //  64*P   forward spectra (32 complex)
    float* U      = Xf    + 64 * P;       //  32*P   stage-1 scratch (16 complex)
    float* eo     = U     + 32 * P;       //  32*P   stage-2 scratch (16 complex)
    float* S1     = U;                    //  aliases U+eo (64*P) for forward FFT
    float* energy = eo    + 32 * P;       // 128*P   [z(32)][o(4)]
    float* e2     = energy + 128 * P;     // 128*P
    float* ampsum = e2    + 128 * P;      //  32*P
    float* gmag   = ampsum + 32 * P;      //  32*P
    float* em     = gmag  + 32 * P;       //   4
    float* saa    = em + 4;               //   4
    float* thr    = saa + 4;              // 128
    float* numb   = thr + 128;            //  16
    float* denb   = numb + 16;            //  16
    unsigned* hist   = (unsigned*)(denb + 16);  // 32768
    unsigned* prefix = hist + 32768;            // 128
    int*      rank   = (int*)(prefix + 128);    // 128

    dim3 b256(256);
    const int iP = (int)P;

    // Downsample + YIQ, DFT matrix, filter bank (+ em).
    k_yiq<<<dim3(96 * P / 256), b256, 0, stream>>>(x, y, yiq);
    k_dft<<<dim3(P / 256), b256, 0, stream>>>(F);
    k_zero<<<1, 64, 0, stream>>>(em, 8);  // em(4) + saa(4)
    k_filters<<<dim3(P / 256), b256, 0, stream>>>(filt, em);

    // g = ifft2(filters).real summed over scales -> sum_aiaj (via WMMA DFTs)
    k_cgemm<false,false><<<dim3(32, 1, 16), b256, 0, stream>>>(filt, iP, 0, 1.0f, nullptr, 0,
                                                  F, 0, iP, -1.0f, U, 2 * iP, 1.0f);
    k_cgemm<true,false><<<dim3(32, 1, 16), b256, 0, stream>>>(F, 0, iP, -1.0f, nullptr, 0,
                                                  U, 2 * iP, iP, 1.0f, eo, 2 * iP, 1.0f / 65536.0f);
    k_red_g<<<dim3(256, 4), b256, 0, stream>>>(eo, saa);

    // Forward FFT of the 32 Y planes (x255 scale folded into stage 1).
    k_cgemm<false,false><<<dim3(32, 1, 32), b256, 0, stream>>>(yiq, 3 * iP, 0, 1.0f, nullptr, 0,
                                                  F, 0, iP, 1.0f, S1, 2 * iP, 255.0f);
    k_cgemm<true,false><<<dim3(32, 1, 32), b256, 0, stream>>>(F, 0, iP, 1.0f, nullptr, 0,
                                                  S1, 2 * iP, iP, 1.0f, Xf, 2 * iP, 1.0f);

    // Per image: filtered inverse FFTs (filter fused into LDS A strip) + energies.
    for (int n = 0; n < 32; ++n) {
        k_cgemm<true,true><<<dim3(32, 1, 16), b256, 0, stream>>>(Xf + (size_t)n * 2 * P, 0, iP, 1.0f,
                                                      filt, iP,
                                                      F, 0, iP, -1.0f, U, 2 * iP, 1.0f);
        k_cgemm<true,false><<<dim3(32, 1, 16), b256, 0, stream>>>(F, 0, iP, -1.0f, nullptr, 0,
                                                      U, 2 * iP, iP, 1.0f, eo, 2 * iP, 1.0f / 65536.0f);
        k_energy<<<dim3(P / 256), b256, 0, stream>>>(eo, energy + (size_t)n * 4 * P,
                                                     e2 + (size_t)n * 4 * P,
                                                     ampsum + (size_t)n * P);
    }

    // Median of e2 per (image, orientation): 4-pass radix select.
    k_ms_init<<<1, 128, 0, stream>>>(prefix, rank);
    for (int shift = 24; shift >= 0; shift -= 8) {
        k_zero_u32<<<dim3(32768 / 256), b256, 0, stream>>>(hist, 32768);
        k_hist<<<dim3(32, 128), b256, 0, stream>>>(e2, prefix, hist, shift);
        k_select<<<1, 128, 0, stream>>>(prefix, rank, hist, shift);
    }
    k_thresh<<<1, 128, 0, stream>>>(prefix, em, saa, thr);

    // Gradient maps + final weighted reduction.
    k_scharr<<<dim3(P / 256, 1, 32), b256, 0, stream>>>(yiq, kern, gmag);
    k_zero<<<1, 64, 0, stream>>>(numb, 32);
    k_fsacc<<<dim3(P / 256, 1, 16), b256, 0, stream>>>(energy, thr, ampsum, gmag, yiq, numb, denb);
    k_finish<<<1, 32, 0, stream>>>(numb, denb, out);
}